// MaterialPropertyPredictor_73547019976733
// MI455X (gfx1250) — compile-verified
//
#include <hip/hip_runtime.h>

typedef float v2f __attribute__((ext_vector_type(2)));
typedef float v8f __attribute__((ext_vector_type(8)));

#define BLOCK  256
#define NWAVES 8
#define G      64

// C(MxN) = act( A(MxK, LDS) * B(KxN, global row-major) + bias ), result -> LDS.
// Each wave computes whole 16x16 tiles with V_WMMA_F32_16X16X4_F32.
// Tile loop trip count is wave-uniform -> EXEC stays all-ones inside WMMA.
__device__ __forceinline__ void gemm_wmma_f32(
    const float* __restrict__ Als, int lda,
    const float* __restrict__ Bg,  int ldb,
    const float* __restrict__ bias,
    float*       __restrict__ Cls, int ldc,
    int M, int N, int K, bool do_relu,
    int wave, int lane)
{
    const int ntiles_n = N >> 4;
    const int ntiles   = (M >> 4) * ntiles_n;
    const int half     = lane >> 4;   // 0: lanes 0-15, 1: lanes 16-31
    const int l16      = lane & 15;

    for (int t = wave; t < ntiles; t += NWAVES) {
        const int mt = (t / ntiles_n) << 4;
        const int nt = (t % ntiles_n) << 4;

        v8f c = {0.f, 0.f, 0.f, 0.f, 0.f, 0.f, 0.f, 0.f};

        // A fragment: lane holds A[mt + l16][k0 + 2*half + {0,1}]  (contiguous pair)
        const float* arow = Als + (mt + l16) * lda + (half << 1);
        // B fragment: lane holds B[k0 + 2*half + {0,1}][nt + l16]
        const float* bcol = Bg + (size_t)(half << 1) * ldb + nt + l16;

        for (int k0 = 0; k0 < K; k0 += 4) {
            v2f a = *(const v2f*)(arow + k0);            // 8B-aligned LDS load
            v2f b;
            b.x = bcol[(size_t)k0 * ldb];
            b.y = bcol[(size_t)(k0 + 1) * ldb];
            c = __builtin_amdgcn_wmma_f32_16x16x4_f32(
                    /*neg_a=*/false, a, /*neg_b=*/false, b,
                    /*c_mod=*/(short)0, c, /*reuse_a=*/false, /*reuse_b=*/false);
        }

        const float bn = bias ? bias[nt + l16] : 0.f;
        #pragma unroll
        for (int v = 0; v < 8; ++v) {
            float x = c[v] + bn;
            if (do_relu) x = fmaxf(x, 0.f);
            // D layout: row = mt + v + 8*half, col = nt + l16
            Cls[(mt + v + (half << 3)) * ldc + nt + l16] = x;
        }
    }
}

__global__ __launch_bounds__(BLOCK)
void MaterialPropertyPredictor_73547019976733_kernel(
    const float* __restrict__ positions, const float* __restrict__ grid_points,
    const float* __restrict__ W1,  const float* __restrict__ b1,
    const float* __restrict__ W2,  const float* __restrict__ b2,
    const float* __restrict__ W3,  const float* __restrict__ b3,
    const float* __restrict__ Wn1, const float* __restrict__ bn1,
    const float* __restrict__ Wn2, const float* __restrict__ bn2,
    const float* __restrict__ Wh1, const float* __restrict__ bh1,
    const float* __restrict__ Wh2, const float* __restrict__ bh2,
    float* __restrict__ out)
{
    // Three 64x256 f32 ping-pong buffers (192 KB) + small state; fits in 320 KB WGP LDS.
    __shared__ float P0[G * 256];        // feat, later y
    __shared__ float P1[G * 256];        // h1, later grid_feats
    __shared__ float P2[G * 256];        // h2, later t = relu(grid@Wn1+bn1)
    __shared__ float s_pos[G * 3];
    __shared__ int   s_win[G];
    __shared__ float s_agg[256];
    __shared__ float s_head[4 * 128];

    const int tid  = threadIdx.x;
    const int lane = tid & 31;
    const int wave = tid >> 5;

    if (tid < G * 3) s_pos[tid] = positions[tid];
    if (tid < G)     s_win[tid] = -1;
    __syncthreads();

    // ---- nearest grid point per position (first 64 rows), winner scatter-max ----
    if (tid < G) {
        const float px = s_pos[tid * 3 + 0];
        const float py = s_pos[tid * 3 + 1];
        const float pz = s_pos[tid * 3 + 2];
        float best = 3.402823466e38f;
        int   bi   = 0;
        for (int j = 0; j < G; ++j) {
            const float dx = px - grid_points[j * 3 + 0];
            const float dy = py - grid_points[j * 3 + 1];
            const float dz = pz - grid_points[j * 3 + 2];
            const float d  = dx * dx + dy * dy + dz * dz;
            if (d < best) { best = d; bi = j; }   // strict '<' keeps first argmin (jnp semantics)
        }
        atomicMax(&s_win[bi], tid);
    }

    // ---- layer 1 (K=3, too skinny for WMMA): h1[64][64] -> P1 ----
    for (int i = tid; i < G * 64; i += BLOCK) {
        const int m = i >> 6, n = i & 63;
        float acc = b1[n];
        acc += s_pos[m * 3 + 0] * W1[0 * 64 + n];
        acc += s_pos[m * 3 + 1] * W1[1 * 64 + n];
        acc += s_pos[m * 3 + 2] * W1[2 * 64 + n];
        P1[m * 64 + n] = fmaxf(acc, 0.f);
    }
    __syncthreads();

    // ---- layer 2: h2[64][128] = relu(h1 @ W2 + b2) -> P2 ----
    gemm_wmma_f32(P1, 64, W2, 128, b2, P2, 128, G, 128, 64, true, wave, lane);
    __syncthreads();

    // ---- layer 3: feat[64][256] = h2 @ W3 + b3 -> P0 ----
    gemm_wmma_f32(P2, 128, W3, 256, b3, P0, 256, G, 256, 128, false, wave, lane);
    __syncthreads();

    // ---- gather: grid_feats[g] = winner[g]>=0 ? feat[winner[g]] : 0 -> P1 ----
    for (int i = tid; i < G * 256; i += BLOCK) {
        const int g = i >> 8, cidx = i & 255;
        const int w = s_win[g];
        P1[i] = (w >= 0) ? P0[w * 256 + cidx] : 0.f;
    }
    __syncthreads();

    // ---- grid net: t = relu(grid_feats @ Wn1 + bn1) -> P2 ----
    gemm_wmma_f32(P1, 256, Wn1, 256, bn1, P2, 256, G, 256, 256, true, wave, lane);
    __syncthreads();

    // ---- y = t @ Wn2 + bn2 -> P0 ----
    gemm_wmma_f32(P2, 256, Wn2, 256, bn2, P0, 256, G, 256, 256, false, wave, lane);
    __syncthreads();

    // ---- agg[c] = mean over 64 grid slots of y[:, c] ----
    if (tid < 256) {
        float s = 0.f;
        for (int g = 0; g < G; ++g) s += P0[g * 256 + tid];
        s_agg[tid] = s * 0.015625f;   // 1/64
    }
    __syncthreads();

    // ---- heads: h1k[k][d] = relu(agg . Wh1[k][:,d] + bh1[k][d]); fold in Wh2 ----
    for (int i = tid; i < 4 * 128; i += BLOCK) {
        const int k = i >> 7, d = i & 127;
        float acc = bh1[k * 128 + d];
        const float* w = Wh1 + (size_t)k * 256 * 128 + d;
        for (int cidx = 0; cidx < 256; ++cidx)
            acc += s_agg[cidx] * w[(size_t)cidx * 128];
        acc = fmaxf(acc, 0.f);
        s_head[i] = acc * Wh2[k * 128 + d];
    }
    __syncthreads();

    if (tid < 4) {
        float s = bh2[tid];
        for (int d = 0; d < 128; ++d) s += s_head[tid * 128 + d];
        out[tid] = s;
    }
}

extern "C" void kernel_launch(void* const* d_in, const int* in_sizes, int n_in,
                              void* d_out, int out_size, void* d_ws, size_t ws_size,
                              hipStream_t stream) {
    (void)in_sizes; (void)n_in; (void)d_ws; (void)ws_size; (void)out_size;
    const float* positions   = (const float*)d_in[0];
    const float* grid_points = (const float*)d_in[1];
    const float* W1  = (const float*)d_in[2];
    const float* b1  = (const float*)d_in[3];
    const float* W2  = (const float*)d_in[4];
    const float* b2  = (const float*)d_in[5];
    const float* W3  = (const float*)d_in[6];
    const float* b3  = (const float*)d_in[7];
    const float* Wn1 = (const float*)d_in[8];
    const float* bn1 = (const float*)d_in[9];
    const float* Wn2 = (const float*)d_in[10];
    const float* bn2 = (const float*)d_in[11];
    const float* Wh1 = (const float*)d_in[12];
    const float* bh1 = (const float*)d_in[13];
    const float* Wh2 = (const float*)d_in[14];
    const float* bh2 = (const float*)d_in[15];

    MaterialPropertyPredictor_73547019976733_kernel<<<1, BLOCK, 0, stream>>>(
        positions, grid_points, W1, b1, W2, b2, W3, b3,
        Wn1, bn1, Wn2, bn2, Wh1, bh1, Wh2, bh2, (float*)d_out);
}